// SplineCNN_70059506532948
// MI455X (gfx1250) — compile-verified
//
#include <hip/hip_runtime.h>
#include <hip/hip_bf16.h>
#include <math.h>

typedef float v2f __attribute__((ext_vector_type(2)));
typedef float v8f __attribute__((ext_vector_type(8)));

#define N_NODES 100000
#define N_EDGES 1600000
#define IN_DIM  500
#define HID_DIM 64
#define N_CLS   16

// Workspace layout (float offsets). Total ~128.4 MB.
#define OFF_XW   0ull          // [N_NODES, 192] = xw0|xw1|xroot
#define OFF_AGG1 19200000ull   // [N_NODES, 64]  layer-1 accumulator, then h
#define OFF_AGG2 25600000ull   // [N_NODES, 16]  layer-2 accumulator
#define OFF_HW   27200000ull   // [N_NODES, 48]  = hw0|hw1|hroot
#define OFF_BT1  32000000ull   // [192, 500]     transposed layer-1 weights
#define OFF_BT2  32096000ull   // [48, 64]       transposed layer-2 weights

// Native CDNA5 f32 atomic add (no return -> STOREcnt), device scope so the
// RMW is performed in GL2 where all WGPs are coherent.
__device__ __forceinline__ void atomic_add_f32(float* p, float v) {
    asm volatile("global_atomic_add_f32 %0, %1, off scope:SCOPE_DEV"
                 :: "v"(p), "v"(v) : "memory");
}

// ---- weight transposes: [K, 64/16] blocks -> N-major [Ntot, K] ----
__global__ void k_prep_bt1(const float* __restrict__ w1, const float* __restrict__ root1,
                           float* __restrict__ bt) {
    int t = blockIdx.x * blockDim.x + threadIdx.x;
    if (t >= 192 * IN_DIM) return;
    int n = t / IN_DIM, k = t - n * IN_DIM;
    const float* s = (n < 128) ? (w1 + (size_t)(n >> 6) * (IN_DIM * HID_DIM)) : root1;
    bt[t] = s[(size_t)k * HID_DIM + (n & 63)];
}

__global__ void k_prep_bt2(const float* __restrict__ w2, const float* __restrict__ root2,
                           float* __restrict__ bt) {
    int t = blockIdx.x * blockDim.x + threadIdx.x;
    if (t >= 48 * HID_DIM) return;
    int n = t / HID_DIM, k = t - n * HID_DIM;
    const float* s = (n < 32) ? (w2 + (size_t)(n >> 4) * (HID_DIM * N_CLS)) : root2;
    bt[t] = s[(size_t)k * N_CLS + (n & 15)];
}

__global__ void k_zero(float* __restrict__ p, int n4) {
    int t = blockIdx.x * blockDim.x + threadIdx.x;
    if (t < n4) ((float4*)p)[t] = make_float4(0.f, 0.f, 0.f, 0.f);
}

// ---- GEMM 1: [100000, 500] x [500, 192] via V_WMMA_F32_16X16X4_F32 ----
// One wave computes a 16x64 tile of C. grid = (6250, 3), block = 32.
__global__ __launch_bounds__(32) void k_gemm1(const float* __restrict__ x,
                                              const float* __restrict__ bt,   // [192, 500]
                                              float* __restrict__ out) {      // [N, 192]
    const int lane = threadIdx.x;
    const int mt = blockIdx.x;
    const int nc = blockIdx.y;                // 64-col chunk
    const int kb = (lane >> 4) << 1;          // 0 or 2
    const float* arow = x  + (size_t)(mt * 16 + (lane & 15)) * IN_DIM + kb;
    const float* brow = bt + (size_t)(nc * 64 + (lane & 15)) * IN_DIM + kb;
    v8f acc0 = {}, acc1 = {}, acc2 = {}, acc3 = {};
#pragma unroll 5
    for (int k0 = 0; k0 < IN_DIM; k0 += 4) {
        v2f a   = *(const v2f*)(arow + k0);
        v2f b0v = *(const v2f*)(brow + k0);
        v2f b1v = *(const v2f*)(brow + 16 * IN_DIM + k0);
        v2f b2v = *(const v2f*)(brow + 32 * IN_DIM + k0);
        v2f b3v = *(const v2f*)(brow + 48 * IN_DIM + k0);
        acc0 = __builtin_amdgcn_wmma_f32_16x16x4_f32(false, a, false, b0v, (short)0, acc0, false, false);
        acc1 = __builtin_amdgcn_wmma_f32_16x16x4_f32(false, a, false, b1v, (short)0, acc1, false, false);
        acc2 = __builtin_amdgcn_wmma_f32_16x16x4_f32(false, a, false, b2v, (short)0, acc2, false, false);
        acc3 = __builtin_amdgcn_wmma_f32_16x16x4_f32(false, a, false, b3v, (short)0, acc3, false, false);
    }
    const int rbase = (lane >> 4) * 8;
    float* o = out + (size_t)(mt * 16) * 192 + nc * 64 + (lane & 15);
#pragma unroll
    for (int r = 0; r < 8; ++r) {
        size_t ro = (size_t)(rbase + r) * 192;
        o[ro] = acc0[r]; o[ro + 16] = acc1[r]; o[ro + 32] = acc2[r]; o[ro + 48] = acc3[r];
    }
}

// ---- edge scatter layer 1: msg = (1-v)*xw0[src] + v*xw1[src] -> atomicAdd agg1[dst] ----
__global__ void k_edge1(const int* __restrict__ ei, const float* __restrict__ attr,
                        const float* __restrict__ xw, float* __restrict__ agg) {
    int t = blockIdx.x * blockDim.x + threadIdx.x;
    if (t >= N_EDGES * 16) return;
    int e = t >> 4;
    int c = (t & 15) << 2;
    float v = attr[e];
    v = fminf(fmaxf(v, 0.0f), 1.0f - 1e-6f);
    int s = ei[e];
    int d = ei[N_EDGES + e];
    float4 a = *(const float4*)(xw + (size_t)s * 192 + c);
    float4 b = *(const float4*)(xw + (size_t)s * 192 + 64 + c);
    float u = 1.0f - v;
    float* q = agg + (size_t)d * 64 + c;
    atomic_add_f32(q + 0, u * a.x + v * b.x);
    atomic_add_f32(q + 1, u * a.y + v * b.y);
    atomic_add_f32(q + 2, u * a.z + v * b.z);
    atomic_add_f32(q + 3, u * a.w + v * b.w);
}

// ---- h = relu(agg1 + xroot + b1), written in place to agg1 ----
__global__ void k_relu(const float* __restrict__ xw, const float* __restrict__ b1,
                       float* __restrict__ agg) {
    int t = blockIdx.x * blockDim.x + threadIdx.x;
    if (t >= N_NODES * 16) return;
    int n = t >> 4;
    int c = (t & 15) << 2;
    float4 a  = *(const float4*)(agg + (size_t)n * 64 + c);
    float4 r  = *(const float4*)(xw + (size_t)n * 192 + 128 + c);
    float4 bb = *(const float4*)(b1 + c);
    float4 h;
    h.x = fmaxf(a.x + r.x + bb.x, 0.f);
    h.y = fmaxf(a.y + r.y + bb.y, 0.f);
    h.z = fmaxf(a.z + r.z + bb.z, 0.f);
    h.w = fmaxf(a.w + r.w + bb.w, 0.f);
    *(float4*)(agg + (size_t)n * 64 + c) = h;
}

// ---- GEMM 2: [100000, 64] x [64, 48], one wave per 16-row tile, all 48 cols ----
__global__ __launch_bounds__(32) void k_gemm2(const float* __restrict__ h,
                                              const float* __restrict__ bt,   // [48, 64]
                                              float* __restrict__ out) {      // [N, 48]
    const int lane = threadIdx.x;
    const int mt = blockIdx.x;
    const int kb = (lane >> 4) << 1;
    const float* arow = h  + (size_t)(mt * 16 + (lane & 15)) * HID_DIM + kb;
    const float* brow = bt + (size_t)(lane & 15) * HID_DIM + kb;
    v8f acc0 = {}, acc1 = {}, acc2 = {};
#pragma unroll
    for (int k0 = 0; k0 < HID_DIM; k0 += 4) {
        v2f a   = *(const v2f*)(arow + k0);
        v2f b0v = *(const v2f*)(brow + k0);
        v2f b1v = *(const v2f*)(brow + 16 * HID_DIM + k0);
        v2f b2v = *(const v2f*)(brow + 32 * HID_DIM + k0);
        acc0 = __builtin_amdgcn_wmma_f32_16x16x4_f32(false, a, false, b0v, (short)0, acc0, false, false);
        acc1 = __builtin_amdgcn_wmma_f32_16x16x4_f32(false, a, false, b1v, (short)0, acc1, false, false);
        acc2 = __builtin_amdgcn_wmma_f32_16x16x4_f32(false, a, false, b2v, (short)0, acc2, false, false);
    }
    const int rbase = (lane >> 4) * 8;
    float* o = out + (size_t)(mt * 16) * 48 + (lane & 15);
#pragma unroll
    for (int r = 0; r < 8; ++r) {
        size_t ro = (size_t)(rbase + r) * 48;
        o[ro] = acc0[r]; o[ro + 16] = acc1[r]; o[ro + 32] = acc2[r];
    }
}

// ---- edge scatter layer 2 (16 channels) ----
__global__ void k_edge2(const int* __restrict__ ei, const float* __restrict__ attr,
                        const float* __restrict__ hw, float* __restrict__ agg) {
    int t = blockIdx.x * blockDim.x + threadIdx.x;
    if (t >= N_EDGES * 4) return;
    int e = t >> 2;
    int c = (t & 3) << 2;
    float v = attr[e];
    v = fminf(fmaxf(v, 0.0f), 1.0f - 1e-6f);
    int s = ei[e];
    int d = ei[N_EDGES + e];
    float4 a = *(const float4*)(hw + (size_t)s * 48 + c);
    float4 b = *(const float4*)(hw + (size_t)s * 48 + 16 + c);
    float u = 1.0f - v;
    float* q = agg + (size_t)d * 16 + c;
    atomic_add_f32(q + 0, u * a.x + v * b.x);
    atomic_add_f32(q + 1, u * a.y + v * b.y);
    atomic_add_f32(q + 2, u * a.z + v * b.z);
    atomic_add_f32(q + 3, u * a.w + v * b.w);
}

// ---- out = log_softmax(agg2 + hroot + b2), 16 classes per node ----
__global__ void k_lsm(const float* __restrict__ agg2, const float* __restrict__ hw,
                      const float* __restrict__ b2, float* __restrict__ out) {
    int n = blockIdx.x * blockDim.x + threadIdx.x;
    if (n >= N_NODES) return;
    float r[16];
    float mx = -3.4e38f;
#pragma unroll
    for (int i = 0; i < 16; ++i) {
        r[i] = agg2[(size_t)n * 16 + i] + hw[(size_t)n * 48 + 32 + i] + b2[i];
        mx = fmaxf(mx, r[i]);
    }
    float s = 0.f;
#pragma unroll
    for (int i = 0; i < 16; ++i) s += __expf(r[i] - mx);
    float ls = __logf(s);
#pragma unroll
    for (int i = 0; i < 16; ++i) out[(size_t)n * 16 + i] = r[i] - mx - ls;
}

extern "C" void kernel_launch(void* const* d_in, const int* in_sizes, int n_in,
                              void* d_out, int out_size, void* d_ws, size_t ws_size,
                              hipStream_t stream) {
    const float* x     = (const float*)d_in[0];
    const int*   ei    = (const int*)d_in[1];
    const float* attr  = (const float*)d_in[2];
    const float* w1    = (const float*)d_in[3];
    const float* root1 = (const float*)d_in[4];
    const float* b1    = (const float*)d_in[5];
    const float* w2    = (const float*)d_in[6];
    const float* root2 = (const float*)d_in[7];
    const float* b2    = (const float*)d_in[8];
    float* ws   = (float*)d_ws;
    float* xw   = ws + OFF_XW;
    float* agg1 = ws + OFF_AGG1;
    float* agg2 = ws + OFF_AGG2;
    float* hw   = ws + OFF_HW;
    float* bt1  = ws + OFF_BT1;
    float* bt2  = ws + OFF_BT2;

    // prep: transpose weights, zero accumulators (agg1 and agg2 are contiguous: 8M floats)
    k_prep_bt1<<<(192 * IN_DIM + 255) / 256, 256, 0, stream>>>(w1, root1, bt1);
    k_prep_bt2<<<(48 * HID_DIM + 255) / 256, 256, 0, stream>>>(w2, root2, bt2);
    {
        int n4 = (N_NODES * HID_DIM + N_NODES * N_CLS) / 4;  // 2,000,000 float4s
        k_zero<<<(n4 + 255) / 256, 256, 0, stream>>>(agg1, n4);
    }
    // layer 1
    k_gemm1<<<dim3(N_NODES / 16, 3), 32, 0, stream>>>(x, bt1, xw);
    k_edge1<<<(N_EDGES * 16) / 256, 256, 0, stream>>>(ei, attr, xw, agg1);
    k_relu<<<(N_NODES * 16) / 256, 256, 0, stream>>>(xw, b1, agg1);
    // layer 2
    k_gemm2<<<N_NODES / 16, 32, 0, stream>>>(agg1, bt2, hw);
    k_edge2<<<(N_EDGES * 4) / 256, 256, 0, stream>>>(ei, attr, hw, agg2);
    k_lsm<<<(N_NODES + 127) / 128, 128, 0, stream>>>(agg2, hw, b2, (float*)d_out);
}